// RL_Deconv_76201309765919
// MI455X (gfx1250) — compile-verified
//
#include <hip/hip_runtime.h>
#include <stdint.h>

#define IMG_H 512
#define IMG_W 512
#define NIMG  4
#define NCH   3
#define NITER 30
#define RLTOL 1e-5
#define TILE  16
// rel/inputs halo: 18x18 ; latent halo: 20x20
#define RHALO (TILE + 2)                     // 18
#define LHALO (TILE + 4)                     // 20
#define RPIX  (RHALO * RHALO)                // 324
#define RELEMS (RPIX * NCH)                  // 972
#define LPIX  (LHALO * LHALO)                // 400
#define LELEMS (LPIX * NCH)                  // 1200
#define PLANE ((size_t)IMG_H * IMG_W * NCH)          // 786432
#define NELEM ((size_t)NIMG * IMG_H * IMG_W * NCH)   // 3145728

__device__ __forceinline__ int reflect_idx(int i, int n) {
    if (i < 0) i = -i;                 // pad<=2 reflect: -1 -> 1, -2 -> 2
    if (i >= n) i = 2 * n - 2 - i;     // 512 -> 510, 513 -> 509
    return i;
}

// Issue per-lane async global->LDS B32 copies of a reflect-padded halo tile.
// Caller must s_wait_asynccnt + barrier before reading the LDS tile.
__device__ __forceinline__ void stage_tile_async(const float* __restrict__ base,
                                                 float* sm, int y0, int x0,
                                                 int halo, int pad, int nelems,
                                                 int tid) {
    uint32_t lds_base = (uint32_t)(uintptr_t)sm;
    for (int e = tid; e < nelems; e += 256) {
        int p  = e / 3;
        int c  = e - p * 3;
        int r  = p / halo;
        int cl = p - r * halo;
        int gy = reflect_idx(y0 - pad + r, IMG_H);
        int gx = reflect_idx(x0 - pad + cl, IMG_W);
        uint64_t ga = (uint64_t)(uintptr_t)(base + ((size_t)(gy * IMG_W + gx)) * NCH + c);
        uint32_t lo = lds_base + (uint32_t)e * 4u;
        asm volatile("global_load_async_to_lds_b32 %0, %1, off"
                     :: "v"(lo), "v"(ga) : "memory");
    }
}

// One fused Richardson-Lucy step for a 16x16x3 output tile:
//   est = conv3x3(latent,k) over an 18x18 halo (evaluated at reflected coords),
//   rel = inputs/est kept in LDS, err = conv3x3(rel,kf) at 16x16,
//   latent *= err (gated by done; latent re-read from LDS, not global),
//   sum(err) accumulated for the convergence mean test.
__global__ __launch_bounds__(256) void k_rl_step(float* __restrict__ latent,
                                                 const float* __restrict__ inputs,
                                                 const float* __restrict__ kern,
                                                 const float* __restrict__ kernf,
                                                 double* __restrict__ sum_buf,
                                                 const int* __restrict__ done) {
    __shared__ float lat[LELEMS];   // 20x20x3 latent halo
    __shared__ float inp[RELEMS];   // 18x18x3 inputs halo
    __shared__ float rel[RELEMS];   // 18x18x3 relative blur
    __shared__ float red[256];

    int tid = threadIdx.x;
    int x0 = blockIdx.x * TILE, y0 = blockIdx.y * TILE, img = blockIdx.z;
    const float* lat_base = latent + (size_t)img * PLANE;
    const float* inp_base = inputs + (size_t)img * PLANE;

    stage_tile_async(lat_base, lat, y0, x0, LHALO, 2, LELEMS, tid);
    stage_tile_async(inp_base, inp, y0, x0, RHALO, 1, RELEMS, tid);
    asm volatile("s_wait_asynccnt 0x0" ::: "memory");
    __syncthreads();

    // kernel weights (uniform -> scalar regs)
    float kw[9][3], kf[9][3];
#pragma unroll
    for (int t = 0; t < 9; ++t)
#pragma unroll
        for (int c = 0; c < 3; ++c) {
            kw[t][c] = kern[t * 3 + c];
            kf[t][c] = kernf[t * 3 + c];
        }

    // Phase 1: rel over the 18x18 halo. For halo pixels, evaluate conv1 at the
    // reflected coordinate so rel(halo) == reflect-pad of full-image rel.
#pragma unroll
    for (int it = 0; it < 2; ++it) {
        int p = tid + it * 256;
        if (p < RPIX) {
            int ly = p / RHALO;
            int lx = p - ly * RHALO;
            int gy = reflect_idx(y0 - 1 + ly, IMG_H);
            int gx = reflect_idx(x0 - 1 + lx, IMG_W);
            int ry = gy - y0 + 2;          // latent-LDS row of (gy)
            int rx = gx - x0 + 2;          // latent-LDS col of (gx)
            float est[3] = {0.f, 0.f, 0.f};
#pragma unroll
            for (int dy = 0; dy < 3; ++dy)
#pragma unroll
                for (int dx = 0; dx < 3; ++dx) {
                    const float* q = &lat[((ry - 1 + dy) * LHALO + (rx - 1 + dx)) * NCH];
#pragma unroll
                    for (int c = 0; c < 3; ++c)
                        est[c] = fmaf(kw[dy * 3 + dx][c], q[c], est[c]);
                }
#pragma unroll
            for (int c = 0; c < 3; ++c)
                rel[p * NCH + c] = inp[p * NCH + c] / est[c];
        }
    }
    __syncthreads();

    // Phase 2: err = conv(rel, kf) at this thread's pixel; update latent.
    int py = tid >> 4, px = tid & 15;
    float err[3] = {0.f, 0.f, 0.f};
#pragma unroll
    for (int dy = 0; dy < 3; ++dy)
#pragma unroll
        for (int dx = 0; dx < 3; ++dx) {
            const float* q = &rel[((py + dy) * RHALO + (px + dx)) * NCH];
#pragma unroll
            for (int c = 0; c < 3; ++c)
                err[c] = fmaf(kf[dy * 3 + dx][c], q[c], err[c]);
        }

    // Current latent value for this pixel is already in the LDS tile
    // (interior element at row py+2, col px+2) -- no global re-read.
    const float* lcur = &lat[((py + 2) * LHALO + (px + 2)) * NCH];
    size_t o = (size_t)img * PLANE + ((size_t)(y0 + py) * IMG_W + (x0 + px)) * NCH;
    if (*done == 0) {
#pragma unroll
        for (int c = 0; c < 3; ++c)
            latent[o + c] = lcur[c] * err[c];
    }

    // Block reduction of sum(err) -> one f64 atomic per block.
    red[tid] = err[0] + err[1] + err[2];
    __syncthreads();
#pragma unroll
    for (int s = 128; s > 0; s >>= 1) {
        if (tid < s) red[tid] += red[tid + s];
        __syncthreads();
    }
    if (tid == 0) atomicAdd(sum_buf, (double)red[0]);
}

// done |= (|1 - mean(err)| < TOL); reset accumulator for next iteration.
__global__ void k_finalize(double* sum_buf, int* done) {
    double mean = *sum_buf / (double)NELEM;
    if (fabs(1.0 - mean) < RLTOL) *done = 1;
    *sum_buf = 0.0;
}

__global__ void k_init(double* sum_buf, int* done) {
    *sum_buf = 0.0;
    *done = 0;
}

extern "C" void kernel_launch(void* const* d_in, const int* in_sizes, int n_in,
                              void* d_out, int out_size, void* d_ws, size_t ws_size,
                              hipStream_t stream) {
    const float* inputs = (const float*)d_in[0];
    const float* kern   = (const float*)d_in[1];  // (3,3,1,3) flat: ky*9+kx*3+c
    const float* kernf  = (const float*)d_in[2];
    float* latent   = (float*)d_out;              // latent lives in d_out
    double* sum_buf = (double*)d_ws;
    int*    done    = (int*)((char*)d_ws + sizeof(double));

    hipMemcpyAsync(latent, inputs, NELEM * sizeof(float),
                   hipMemcpyDeviceToDevice, stream);
    k_init<<<1, 1, 0, stream>>>(sum_buf, done);

    dim3 grid(IMG_W / TILE, IMG_H / TILE, NIMG);   // 32 x 32 x 4
    for (int it = 0; it < NITER; ++it) {
        k_rl_step<<<grid, 256, 0, stream>>>(latent, inputs, kern, kernf,
                                            sum_buf, done);
        k_finalize<<<1, 1, 0, stream>>>(sum_buf, done);
    }
}